// ModelNew_80908593922582
// MI455X (gfx1250) — compile-verified
//
#include <hip/hip_runtime.h>

typedef float v8f __attribute__((ext_vector_type(8)));
typedef float v4f __attribute__((ext_vector_type(4)));
typedef float v2f __attribute__((ext_vector_type(2)));
typedef unsigned u32x4 __attribute__((ext_vector_type(4)));
typedef int i32x4 __attribute__((ext_vector_type(4)));
typedef int i32x8 __attribute__((ext_vector_type(8)));

#define XS_CI_STRIDE 528   // 4*132 dwords: TDM pads 4 dwords per 128 -> 2*528 % 64 == 32
#define XS_ID_STRIDE 132   // 4*32 + 4 pad dwords, stays 16B aligned
#define TDIM_BIG (1 << 20) // tensor dims >= tile dims -> no OOB clamping path

__device__ __forceinline__ unsigned fkey(float f) {
  unsigned b = __float_as_uint(f);
  return (b & 0x80000000u) ? ~b : (b | 0x80000000u);
}
__device__ __forceinline__ float funkey(unsigned k) {
  unsigned b = (k & 0x80000000u) ? (k & 0x7fffffffu) : ~k;
  return __uint_as_float(b);
}

// One workgroup per (n, pd, ph). Computes the 6x6x60 slab of conv-transpose
// outputs for all 32 channels via f32 WMMA, max-folds into 6^3 windows,
// channel-sums, writes 10 outputs.
__global__ __launch_bounds__(256)
void convT_pool_sum(const float* __restrict__ xg,   // (8,16,32,32,32)
                    const float* __restrict__ wg,   // (16,32,3,3,3)
                    const float* __restrict__ bg,   // (32)
                    float* __restrict__ out) {      // (8,1,10,10,10)
  __shared__ float    xs[16 * XS_CI_STRIDE];  // 8448 dwords: x patch (TDM target)
  __shared__ float    bs[27 * 4 * 128];       // 13824 dwords: permuted weights
  __shared__ unsigned mx[320];                // running max keys [co][pw]

  const int ph   = blockIdx.x;
  const int pd   = blockIdx.y;
  const int n    = blockIdx.z;
  const int tid  = threadIdx.x;
  const int lane = tid & 31;
  const int wv   = tid >> 5;

  // ---- kick off TDM: 4D tile (iw32, ih4, id4, ci16) global -> LDS ----
  // D# per CDNA5 ISA sect. 8.3-8.6. pad: 4 dwords every 128 dwords
  // -> LDS layout xs[ci*528 + id*132 + ih*32 + iw]. TDM ignores EXEC; only
  // wave 0 reaches this block, single issue.
  if (tid == 0) {
    const unsigned lds_x = (unsigned)(unsigned long long)(void*)xs;
    const unsigned long long ga = (unsigned long long)(const void*)(xg +
        ((((size_t)n * 16) * 32 + 3 * pd) * 32 + 3 * ph) * 32);
    u32x4 g0;
    g0[0] = 1u;                                   // count=1, user descriptor
    g0[1] = lds_x;                                // lds_addr (bytes)
    g0[2] = (unsigned)ga;                         // global_addr[31:0]
    g0[3] = (unsigned)(ga >> 32) | (2u << 30);    // global_addr[56:32] | type=2
    i32x8 g1;
    g1[0] = (2 << 16) | (1 << 20) | (6 << 22) | (3 << 25); // 4B, pad_en, 128dw, 4dw
    g1[1] = (int)((TDIM_BIG & 0xffffu) << 16);             // tensor_dim0 lo16
    g1[2] = (int)(((TDIM_BIG >> 16) & 0xffffu) | ((TDIM_BIG & 0xffffu) << 16));
    g1[3] = (int)(((TDIM_BIG >> 16) & 0xffffu) | (32u << 16));   // tile_dim0=32
    g1[4] = 4 | (4 << 16);                        // tile_dim1=4 (ih), tile_dim2=4 (id)
    g1[5] = 32;                                   // tensor_dim0_stride (ih step)
    g1[6] = (int)(1024u << 16);                   // tensor_dim1_stride lo16 (id step)
    g1[7] = 0;                                    // tensor_dim1_stride hi
    i32x4 g2;
    g2[0] = TDIM_BIG;                             // tensor_dim2
    g2[1] = TDIM_BIG;                             // tensor_dim3
    g2[2] = 32768;                                // tensor_dim2_stride (ci step)
    g2[3] = (16 << 16);                           // tile_dim3=16 (ci)
    i32x4 g3;
    g3[0] = 0;                                    // tensor_dim3_stride lo
    g3[1] = (1 << 16);                            // tensor_dim4=1
    g3[2] = 0;                                    // tile_dim4=0 (unused)
    g3[3] = 0;
#if defined(__clang_major__) && __clang_major__ >= 23
    i32x8 gz = {0, 0, 0, 0, 0, 0, 0, 0};
    __builtin_amdgcn_tensor_load_to_lds(g0, g1, g2, g3, gz, 0);
#else
    __builtin_amdgcn_tensor_load_to_lds(g0, g1, g2, g3, 0);
#endif
  }

  // ---- stage weights permuted for per-lane B fragments (overlaps TDM) ----
  // layout: bs[((tap*4+q)*128) + lane*4 + half*2 + j], ci = 4q + (lane>>4)*2 + j,
  //         co = half*16 + (lane&15)   -> one ds_load_b128 per (tap,q)
  for (int e = tid; e < 27 * 4 * 128; e += 256) {
    const int j    = e & 1;
    const int half = (e >> 1) & 1;
    const int ln   = (e >> 2) & 31;
    const int q    = (e >> 7) & 3;
    const int tap  = e >> 9;
    const int ci   = 4 * q + ((ln >> 4) << 1) + j;
    const int co   = half * 16 + (ln & 15);
    bs[e] = wg[(ci * 32 + co) * 27 + tap];
  }
  // ---- init max keys (key 0 == minimum) ----
  mx[tid % 320] = 0u;
  if (tid < 64) mx[tid + 256] = 0u;

  __builtin_amdgcn_s_wait_tensorcnt(0);   // no-op for waves that issued nothing
  __syncthreads();

  // ---- belt-and-braces x staging into the same padded layout (coalesced) ----
  // Rewrites the TDM-filled bytes; guarantees correctness independent of the
  // descriptor's LDS-write semantics.
  {
    const int w4 = tid & 7;        // float4 column
    int row = tid >> 3;            // (ci,id,ih) row, 32 rows / iter
    #pragma unroll
    for (int it = 0; it < 8; ++it, row += 32) {
      const int ci = row >> 4, id = (row >> 2) & 3, ih = row & 3;
      const float4 v = *(const float4*)(xg +
          ((((size_t)(n * 16 + ci) * 32 + (3 * pd + id)) * 32 + (3 * ph + ih)) * 32 + w4 * 4));
      *(float4*)&xs[ci * XS_CI_STRIDE + id * XS_ID_STRIDE + ih * 32 + w4 * 4] = v;
    }
  }
  __syncthreads();

  const float bias0 = bg[lane & 15];
  const float bias1 = bg[(lane & 15) + 16];

  // jobs: 8 parity classes x 17 M-tiles = 136, round-robin over 8 waves (17 each)
  for (int job = wv; job < 136; job += 8) {
    const int cls  = job / 17;
    const int tile = job - cls * 17;
    const int a_d = (cls >> 2) & 1, a_h = (cls >> 1) & 1, a_w = cls & 1;

    // per-lane A base address (M = lane&15; lanes>=16 carry K=2,3 -> ci offset 2)
    int mA = tile * 16 + (lane & 15);
    mA = mA < 270 ? mA : 269;                 // tail rows: harmless clamp
    const int u_w = mA % 30;
    const int t2  = mA / 30;
    const int u_h = t2 % 3, u_d = t2 / 3;
    const int abase = ((lane >> 4) << 1) * XS_CI_STRIDE
                    + u_d * XS_ID_STRIDE + u_h * 32 + u_w;

    v8f c0, c1;
    #pragma unroll
    for (int r = 0; r < 8; ++r) { c0[r] = bias0; c1[r] = bias1; }

    const int kd0 = a_d ? 0 : 1, nkd = a_d ? 2 : 1;
    const int kh0 = a_h ? 0 : 1, nkh = a_h ? 2 : 1;
    const int kw0 = a_w ? 0 : 1, nkw = a_w ? 2 : 1;
    for (int ikd = 0; ikd < nkd; ++ikd) {
      const int kd = kd0 + ikd * 2;
      const int offd = (a_d && kd == 0) ? XS_ID_STRIDE : 0;  // odd od, kd=0 -> id=u+1
      for (int ikh = 0; ikh < nkh; ++ikh) {
        const int kh = kh0 + ikh * 2;
        const int offh = (a_h && kh == 0) ? 32 : 0;
        for (int ikw = 0; ikw < nkw; ++ikw) {
          const int kw = kw0 + ikw * 2;
          const int offw = (a_w && kw == 0) ? 1 : 0;
          const int tap  = (kd * 3 + kh) * 3 + kw;
          const int aoff = abase + offd + offh + offw;
          const float* bp = &bs[(tap * 4) * 128 + (lane << 2)];
          #pragma unroll
          for (int q = 0; q < 4; ++q) {
            v2f a;
            a.x = xs[aoff + q * 4 * XS_CI_STRIDE];
            a.y = xs[aoff + q * 4 * XS_CI_STRIDE + XS_CI_STRIDE];
            const v4f bb = *(const v4f*)(bp + q * 128);
            const v2f b0 = bb.xy;
            const v2f b1 = bb.zw;
            // D = A(16x4 f32) x B(4x16 f32) + C : exact fp32 on matrix pipes
            c0 = __builtin_amdgcn_wmma_f32_16x16x4_f32(false, a, false, b0,
                                                       (short)0, c0, false, false);
            c1 = __builtin_amdgcn_wmma_f32_16x16x4_f32(false, a, false, b1,
                                                       (short)0, c1, false, false);
          }
        }
      }
    }

    // fold D into per-(co,pw) running max (D row M = r (+8 for lanes>=16))
    #pragma unroll
    for (int r = 0; r < 8; ++r) {
      const int mrow = r + ((lane >> 4) << 3);
      const int m = tile * 16 + mrow;
      if (m < 270) {
        const int uw = m % 30;
        const int pw = (2 * uw + a_w) / 6;
        const int co = lane & 15;
        atomicMax(&mx[co * 10 + pw], fkey(c0[r]));          // ds_max_u32
        atomicMax(&mx[(co + 16) * 10 + pw], fkey(c1[r]));
      }
    }
  }
  __syncthreads();

  // ---- channel sum, write 10 outputs ----
  if (tid < 10) {
    float s = 0.f;
    #pragma unroll
    for (int co = 0; co < 32; ++co) s += funkey(mx[co * 10 + tid]);
    out[(size_t)n * 1000 + pd * 100 + ph * 10 + tid] = s;
  }
}

extern "C" void kernel_launch(void* const* d_in, const int* in_sizes, int n_in,
                              void* d_out, int out_size, void* d_ws, size_t ws_size,
                              hipStream_t stream) {
  (void)in_sizes; (void)n_in; (void)out_size; (void)d_ws; (void)ws_size;
  const float* x = (const float*)d_in[0];
  const float* w = (const float*)d_in[1];
  const float* b = (const float*)d_in[2];
  float* out = (float*)d_out;
  dim3 grid(10, 10, 8);   // (ph, pd, n)
  convT_pool_sum<<<grid, dim3(256, 1, 1), 0, stream>>>(x, w, b, out);
}